// Topo_Signature_Layer_1941325218289
// MI455X (gfx1250) — compile-verified
//
#include <hip/hip_runtime.h>
#include <hip/hip_bf16.h>
#include <math.h>

typedef __attribute__((ext_vector_type(2))) float v2f;
typedef __attribute__((ext_vector_type(8))) float v8f;

#define NUNITS   64
#define TILES_N  4      // 4 unit-tiles of 16 -> 64 units
#define WAVES    4
#define BLOCK    128    // 4 wave32

// Compute the A-matrix fragment (16x4 f32) for a 16-point tile.
// Lanes 0-15 carry (x0^2, x0) as K=0,1 for point base+col;
// lanes 16-31 carry (xe^2, xe) as K=2,3 for the same point.
__device__ __forceinline__ v2f make_A(const float* __restrict__ X, int pc, bool hi)
{
    const float rot = 0.70710678118654752f;      // cos(45) = sin(45)
    const float lnC = 0.00693147180559945f;      // 0.01 * ln(2) (v_log_f32 = log2)
    const float c2  = 0.05605170185988091f;      // 0.01 - 0.01*ln(0.01)
    v2f xv = *(const v2f*)(X + 2 * (long long)pc);
    float b  = xv[0];
    float dd = xv[1];
    float x0 = rot * (b + dd);
    float x1 = rot * (dd - b);                   // > 0 since pers > 0
    float x1t = lnC * __builtin_amdgcn_logf(x1) + c2;
    float xe  = (x0 >= 0.01f) ? x1 : x1t;        // point-level condition
    float f   = hi ? xe : x0;
    return (v2f){f * f, f};
}

// One block per diagram. Each wave takes every 4th 16-point tile against all
// four 16-unit tiles via V_WMMA_F32_16X16X4_F32; exp2 of the result is
// accumulated into v8f accumulators (8 independent adds, no serial chain).
__global__ __launch_bounds__(BLOCK)
void topo_sig_kernel(const float* __restrict__ X,        // (N,2) [birth, death]
                     const float* __restrict__ mu0,
                     const float* __restrict__ log_mu1,
                     const float* __restrict__ log_s0,
                     const float* __restrict__ log_s1,
                     const int*   __restrict__ slices,   // (ndiag,2) int32 or int64
                     float* __restrict__ out,            // (ndiag,64)
                     int n_points)
{
    const int d    = blockIdx.x;
    const int tid  = threadIdx.x;
    const int wave = __builtin_amdgcn_readfirstlane(tid >> 5); // wave-uniform -> SALU
    const int lane = tid & 31;
    const int col  = lane & 15;
    const bool hi  = lane >= 16;

    // --- slice bounds (probe int32 vs int64 layout; uniform, int32 math) ---
    int s32, e32;
    {
        int w1 = slices[1], w3 = slices[3];
        if (w1 == 0 && w3 == 0) {                // int64 layout (high words zero)
            const long long* s64 = (const long long*)slices;
            s32 = (int)s64[2 * d]; e32 = (int)s64[2 * d + 1];
        } else {
            s32 = slices[2 * d]; e32 = slices[2 * d + 1];
        }
    }

    const float L = 1.44269504088896340736f;     // log2(e), folded into B and C

    // --- per-lane unit parameters: B (2 VGPRs/tile) and C constant ---
    // B 4x16 f32: lanes 0-15 hold K=0 (v.x), K=1 (v.y) for unit = t*16+col;
    //             lanes 16-31 hold K=2 (v.x), K=3 (v.y).
    v2f  Bm[TILES_N];
    float Cc[TILES_N];
#pragma unroll
    for (int t = 0; t < TILES_N; ++t) {
        int u = t * 16 + col;
        float m0  = mu0[u];
        float m1  = expf(log_mu1[u]);
        float s0  = expf(log_s0[u]);
        float s1  = expf(log_s1[u]);
        float s0q = s0 * s0, s1q = s1 * s1;
        float b0 = hi ? (-L * s1q)            : (-L * s0q);
        float b1 = hi ? (2.0f * L * s1q * m1) : (2.0f * L * s0q * m0);
        Bm[t] = (v2f){b0, b1};
        Cc[t] = -L * (s0q * m0 * m0 + s1q * m1 * m1);
    }

    const int count = e32 - s32;
    const int nfull = count >> 4;                // full 16-point tiles
    const bool haspartial = (count & 15) != 0;

    v8f acc[TILES_N];
#pragma unroll
    for (int t = 0; t < TILES_N; ++t)
#pragma unroll
        for (int r = 0; r < 8; ++r) acc[t][r] = 0.0f;

    // --- main loop: full tiles only, no masking, no divergence ---
    for (int i = wave; i < nfull; i += WAVES) {
        int base = s32 + (i << 4);
        v2f A = make_A(X, base + col, hi);       // in-range: full tile
#pragma unroll
        for (int t = 0; t < TILES_N; ++t) {
            v8f c;
#pragma unroll
            for (int r = 0; r < 8; ++r) c[r] = Cc[t];
            v8f dm = __builtin_amdgcn_wmma_f32_16x16x4_f32(
                false, A, false, Bm[t], (short)0, c, false, false);
#pragma unroll
            for (int r = 0; r < 8; ++r)
                acc[t][r] += __builtin_amdgcn_exp2f(dm[r]);
        }
    }

    // --- epilogue: single ragged tile, one wave, row-masked after exp ---
    if (haspartial && ((nfull & (WAVES - 1)) == wave)) {
        int base = s32 + (nfull << 4);
        int p  = base + col;
        int pc = p < (n_points - 1) ? p : (n_points - 1);
        v2f A = make_A(X, pc, hi);
        int rowbase = base + (hi ? 8 : 0);       // D rows: M = r (+8 for hi lanes)
#pragma unroll
        for (int t = 0; t < TILES_N; ++t) {
            v8f c;
#pragma unroll
            for (int r = 0; r < 8; ++r) c[r] = Cc[t];
            v8f dm = __builtin_amdgcn_wmma_f32_16x16x4_f32(
                false, A, false, Bm[t], (short)0, c, false, false);
#pragma unroll
            for (int r = 0; r < 8; ++r) {
                float v = __builtin_amdgcn_exp2f(dm[r]);
                acc[t][r] += ((rowbase + r) < e32) ? v : 0.0f;
            }
        }
    }

    // --- reduce: 8->1 tree, rows 0-7 + rows 8-15 via shfl_xor, then waves ---
    __shared__ float part[WAVES][NUNITS];
#pragma unroll
    for (int t = 0; t < TILES_N; ++t) {
        v8f a = acc[t];
        float v = ((a[0] + a[1]) + (a[2] + a[3])) +
                  ((a[4] + a[5]) + (a[6] + a[7]));
        v += __shfl_xor(v, 16, 32);
        if (!hi) part[wave][t * 16 + col] = v;
    }
    __syncthreads();
    if (tid < NUNITS) {
        float r = part[0][tid] + part[1][tid] + part[2][tid] + part[3][tid];
        out[(long long)d * NUNITS + tid] = r;
    }
}

extern "C" void kernel_launch(void* const* d_in, const int* in_sizes, int n_in,
                              void* d_out, int out_size, void* d_ws, size_t ws_size,
                              hipStream_t stream) {
    const float* X    = (const float*)d_in[0];   // X_persis (N,2)
    const float* mu0  = (const float*)d_in[1];
    const float* lmu1 = (const float*)d_in[2];
    const float* ls0  = (const float*)d_in[3];
    const float* ls1  = (const float*)d_in[4];
    const int*   sl   = (const int*)d_in[5];     // diagram_slices (ndiag,2)
    float* out = (float*)d_out;

    int n_points = in_sizes[0] / 2;
    int ndiag    = in_sizes[5] / 2;

    topo_sig_kernel<<<ndiag, BLOCK, 0, stream>>>(X, mu0, lmu1, ls0, ls1, sl,
                                                 out, n_points);
}